// DeformableTransformer_86105504350451
// MI455X (gfx1250) — compile-verified
//
#include <hip/hip_runtime.h>
#include <hip/hip_bf16.h>
#include <math.h>

// ---------------------------------------------------------------------------
// Deformable Transformer (Deformable-DETR style) for MI455X / gfx1250.
// Dense linear layers: LDS-tiled, double-buffered WMMA f32 GEMM fed by
// async global->LDS copies (GLOBAL_LOAD_ASYNC_TO_LDS_B128, ASYNCcnt).
// Sampling / softmax / LN are VALU kernels.
// ---------------------------------------------------------------------------

typedef __attribute__((ext_vector_type(2))) float v2f;
typedef __attribute__((ext_vector_type(8))) float v8f;

#define D_MODEL 256
#define NHEAD 8
#define DHEAD 32
#define NLVL 4
#define NPT 4
#define DFF_ 2048
#define NQ_ 900
#define BS_ 2
#define SUMHW 17821
#define ENC_TOK (BS_ * SUMHW)   // 35642
#define DEC_TOK (BS_ * NQ_)     // 1800
#define FFN_CHUNK 4096

// LDS tile geometry for the GEMM
#define KC 32          // K chunk
#define MT 32          // M tile per block
#define NT 64          // N tile per block
#define AST 36         // A LDS row stride (floats): 36*4=144B, 16B aligned rows
#define BST 80         // B LDS row stride (floats): 80*4=320B, 16B aligned rows

// ---------------------------------------------------------------------------
// Async global -> LDS copy (16 bytes per lane), CDNA5 ASYNCcnt path.
// ---------------------------------------------------------------------------
__device__ __forceinline__ void async_copy_b128(unsigned lds_addr,
                                                const void* gaddr) {
  asm volatile("global_load_async_to_lds_b128 %0, %1, off" ::"v"(lds_addr),
               "v"((unsigned long long)(size_t)gaddr)
               : "memory");
}
__device__ __forceinline__ void wait_async0() {
  asm volatile("s_wait_asynccnt 0x0" ::: "memory");
}

// ---------------------------------------------------------------------------
// WMMA f32 GEMM:  C[M,N] = act(A[M,K] @ B[K,N] + bias[N])
// Block = 256 threads = 8 waves -> 32x64 C tile (waves 2x4, 16x16 each).
// K consumed in 32-chunks staged in LDS via async copies, double buffered.
// Requires N % 64 == 0 and K % 32 == 0 (true here: N=128/256/512/2048,
// K=256/512/2048).
// ---------------------------------------------------------------------------
template <int RELU>
__global__ __launch_bounds__(256) void wmma_gemm_bias(
    const float* __restrict__ A, const float* __restrict__ B,
    const float* __restrict__ bias, float* __restrict__ C,
    int M, int N, int K) {
  __shared__ float As[2][MT * AST];
  __shared__ float Bs[2][KC * BST];

  const int t = threadIdx.x;
  const int lane = t & 31;
  const int wave = t >> 5;
  const int m0 = blockIdx.x * MT;
  const int n0 = blockIdx.y * NT;

  // ---- staging decomposition (all 256 threads) ----
  // A chunk: MT x KC floats = 1024 -> 256 lanes x 1 float4
  const int a_row = t >> 3;          // 0..31
  const int a_col = (t & 7) * 4;     // 0..28
  int a_grow = m0 + a_row;
  if (a_grow >= M) a_grow = M - 1;   // clamped rows only feed clamped outputs
  // B chunk: KC x NT floats = 2048 -> 256 lanes x 2 float4
  const int b_row = t >> 4;          // 0..15 (and +16)
  const int b_col = (t & 15) * 4;    // 0..60

  // ---- compute decomposition ----
  const int mw = (wave >> 2) * 16;   // 0 or 16
  const int nw = (wave & 3) * 16;    // 0..48
  const int half = lane >> 4;        // 0 or 1
  const int l16 = lane & 15;
  const int col = n0 + nw + l16;

  auto stage = [&](int buf, int k0) {
    async_copy_b128((unsigned)(size_t)&As[buf][a_row * AST + a_col],
                    A + (size_t)a_grow * K + k0 + a_col);
    async_copy_b128((unsigned)(size_t)&Bs[buf][b_row * BST + b_col],
                    B + (size_t)(k0 + b_row) * N + n0 + b_col);
    async_copy_b128((unsigned)(size_t)&Bs[buf][(b_row + 16) * BST + b_col],
                    B + (size_t)(k0 + b_row + 16) * N + n0 + b_col);
  };

  v8f acc = {0.f, 0.f, 0.f, 0.f, 0.f, 0.f, 0.f, 0.f};

  stage(0, 0);
  for (int kc = 0, it = 0; kc < K; kc += KC, ++it) {
    const int cur = it & 1;
    wait_async0();          // our async copies for buffer `cur` have landed
    __syncthreads();        // everyone's copies landed; prev compute done
    if (kc + KC < K) stage(cur ^ 1, kc + KC);
    const float* __restrict__ Ab = &As[cur][0];
    const float* __restrict__ Bb = &Bs[cur][0];
#pragma unroll
    for (int kk = 0; kk < KC; kk += 4) {
      v2f a, b;
      const int ka = kk + 2 * half;
      a.x = Ab[(mw + l16) * AST + ka];
      a.y = Ab[(mw + l16) * AST + ka + 1];
      b.x = Bb[ka * BST + nw + l16];
      b.y = Bb[(ka + 1) * BST + nw + l16];
      acc = __builtin_amdgcn_wmma_f32_16x16x4_f32(false, a, false, b, (short)0,
                                                  acc, false, false);
    }
    __syncthreads();        // done reading `cur` before it is restaged
  }

  const float bz = bias[col];
#pragma unroll
  for (int r = 0; r < 8; ++r) {
    const int m = m0 + mw + r + 8 * half;
    if (m < M) {
      float v = acc[r] + bz;
      if (RELU) v = fmaxf(v, 0.f);
      C[(size_t)m * N + col] = v;
    }
  }
}

// ---------------------------------------------------------------------------
// LayerNorm over D=256, optional residual:  y = LN(x (+ res)) * g + b
// ---------------------------------------------------------------------------
__global__ __launch_bounds__(256) void ln_res_kernel(
    const float* __restrict__ x, const float* __restrict__ res,
    const float* __restrict__ g, const float* __restrict__ bta,
    float* __restrict__ y) {
  const int row = blockIdx.x;
  const int t = threadIdx.x;
  float v = x[(size_t)row * D_MODEL + t];
  if (res) v += res[(size_t)row * D_MODEL + t];
  __shared__ float red[256];
  red[t] = v;
  __syncthreads();
  for (int s = 128; s > 0; s >>= 1) {
    if (t < s) red[t] += red[t + s];
    __syncthreads();
  }
  const float mean = red[0] * (1.f / D_MODEL);
  __syncthreads();
  const float dv = v - mean;
  red[t] = dv * dv;
  __syncthreads();
  for (int s = 128; s > 0; s >>= 1) {
    if (t < s) red[t] += red[t + s];
    __syncthreads();
  }
  const float var = red[0] * (1.f / D_MODEL);
  y[(size_t)row * D_MODEL + t] = dv * rsqrtf(var + 1e-5f) * g[t] + bta[t];
}

__global__ void add_kernel(const float* __restrict__ a,
                           const float* __restrict__ b, float* __restrict__ c,
                           size_t n) {
  size_t i = (size_t)blockIdx.x * blockDim.x + threadIdx.x;
  if (i < n) c[i] = a[i] + b[i];
}

__global__ void sigmoid_kernel(const float* __restrict__ x,
                               float* __restrict__ y, int n) {
  int i = blockIdx.x * blockDim.x + threadIdx.x;
  if (i < n) y[i] = 1.f / (1.f + expf(-x[i]));
}

__global__ void bcast_tgt_kernel(const float* __restrict__ te,
                                 float* __restrict__ tgt) {
  const int row = blockIdx.x;  // 0..DEC_TOK-1
  const int t = threadIdx.x;
  tgt[(size_t)row * D_MODEL + t] = te[(size_t)(row % NQ_) * D_MODEL + t];
}

// softmax over the 32 (level,point) attention logits per (b,q,h), in place
__global__ void softmax32_kernel(float* __restrict__ aw, int total) {
  int i = blockIdx.x * blockDim.x + threadIdx.x;
  if (i >= total) return;
  float* p = aw + (size_t)i * 32;
  float m = p[0];
#pragma unroll
  for (int j = 1; j < 32; ++j) m = fmaxf(m, p[j]);
  float e[32];
  float s = 0.f;
#pragma unroll
  for (int j = 0; j < 32; ++j) {
    e[j] = expf(p[j] - m);
    s += e[j];
  }
  const float inv = 1.f / s;
#pragma unroll
  for (int j = 0; j < 32; ++j) p[j] = e[j] * inv;
}

// encoder reference points rp[b, tok, {x,y}] (before per-level vr scaling)
__global__ void enc_ref_kernel(const float* __restrict__ vr,
                               float* __restrict__ rp) {
  const int i = blockIdx.x * blockDim.x + threadIdx.x;
  if (i >= BS_ * SUMHW) return;
  const int b = i / SUMHW, tok = i % SUMHW;
  const int Hs[4] = {100, 50, 25, 13};
  const int Ws[4] = {134, 67, 34, 17};
  const int St[4] = {0, 13400, 16750, 17600};
  int lvl = 3;
  for (int l = 0; l < 4; ++l) {
    if (tok < St[l] + Hs[l] * Ws[l]) {
      lvl = l;
      break;
    }
  }
  const int loc = tok - St[lvl];
  const int yi = loc / Ws[lvl];
  const int xi = loc % Ws[lvl];
  rp[(size_t)i * 2 + 0] = (xi + 0.5f) / (vr[(b * NLVL + lvl) * 2 + 0] * Ws[lvl]);
  rp[(size_t)i * 2 + 1] = (yi + 0.5f) / (vr[(b * NLVL + lvl) * 2 + 1] * Hs[lvl]);
}

// sine positional embedding of ref_in[:, :, 0] (4 coords -> 4x128 = 512)
__global__ __launch_bounds__(512) void sine_embed_kernel(
    const float* __restrict__ refp, const float* __restrict__ vr,
    float* __restrict__ qse) {
  const int idx = blockIdx.x;  // b*NQ + q
  const int b = idx / NQ_;
  const int c = threadIdx.x;  // 0..511
  const int g = c >> 7, j = c & 127;
  const int dims[4] = {1, 0, 2, 3};
  const int dim = dims[g];
  const float vrv = vr[(b * NLVL + 0) * 2 + (dim & 1)];
  const float x = refp[(size_t)idx * 4 + dim] * vrv;
  const float dim_t = powf(10000.f, (float)(2 * (j >> 1)) * (1.f / 128.f));
  const float e = x * 6.283185307179586f / dim_t;
  qse[(size_t)idx * 512 + c] = (j & 1) ? cosf(e) : sinf(e);
}

// ---------------------------------------------------------------------------
// Deformable attention bilinear sampling.
// Block per (b*Lq + q), 256 threads -> thread = (head h = t>>5, dim d = t&31).
// ---------------------------------------------------------------------------
__global__ __launch_bounds__(256) void msda_sample_kernel(
    const float* __restrict__ value, const float* __restrict__ off,
    const float* __restrict__ aw, const float* __restrict__ refb,
    const float* __restrict__ vr, float* __restrict__ out, int Lq,
    int is_enc) {
  const int idx = blockIdx.x;
  const int b = idx / Lq;
  const int t = threadIdx.x;
  const int h = t >> 5, d = t & 31;
  const int Hs[4] = {100, 50, 25, 13};
  const int Ws[4] = {134, 67, 34, 17};
  const int St[4] = {0, 13400, 16750, 17600};
  float acc = 0.f;
  for (int lvl = 0; lvl < NLVL; ++lvl) {
    const int H = Hs[lvl], W = Ws[lvl], s0 = St[lvl];
    const float vrx = vr[(b * NLVL + lvl) * 2 + 0];
    const float vry = vr[(b * NLVL + lvl) * 2 + 1];
    float rx, ry, rw = 0.f, rh = 0.f;
    if (is_enc) {
      rx = refb[(size_t)idx * 2 + 0] * vrx;
      ry = refb[(size_t)idx * 2 + 1] * vry;
    } else {
      rx = refb[(size_t)idx * 4 + 0] * vrx;
      ry = refb[(size_t)idx * 4 + 1] * vry;
      rw = refb[(size_t)idx * 4 + 2] * vrx;
      rh = refb[(size_t)idx * 4 + 3] * vry;
    }
    for (int p = 0; p < NPT; ++p) {
      const size_t obase =
          ((((size_t)idx * NHEAD + h) * NLVL + lvl) * NPT + p) * 2;
      const float ox = off[obase + 0];
      const float oy = off[obase + 1];
      const float w_aw =
          aw[((size_t)idx * NHEAD + h) * (NLVL * NPT) + lvl * NPT + p];
      float lx, ly;
      if (is_enc) {
        lx = rx + ox / (float)W;
        ly = ry + oy / (float)H;
      } else {
        lx = rx + ox * (0.5f / NPT) * rw;
        ly = ry + oy * (0.5f / NPT) * rh;
      }
      const float px = lx * W - 0.5f;
      const float py = ly * H - 0.5f;
      const float x0 = floorf(px);
      const float y0 = floorf(py);
#pragma unroll
      for (int c = 0; c < 4; ++c) {
        const float xi = x0 + (float)(c & 1);
        const float yi = y0 + (float)(c >> 1);
        if (xi >= 0.f && xi < (float)W && yi >= 0.f && yi < (float)H) {
          const float wgt = (1.f - fabsf(px - xi)) * (1.f - fabsf(py - yi));
          const int ci = (int)yi * W + (int)xi;
          const float gv =
              value[((size_t)(b * SUMHW + s0 + ci)) * D_MODEL + h * DHEAD + d];
          acc += gv * wgt * w_aw;
        }
      }
    }
  }
  out[(size_t)idx * D_MODEL + t] = acc;
}

// ---------------------------------------------------------------------------
// Decoder self-attention: one block per (b,h,q), 128 threads, Nk<=900.
// ---------------------------------------------------------------------------
__global__ __launch_bounds__(128) void mha_attn_kernel(
    const float* __restrict__ qh, const float* __restrict__ kh,
    const float* __restrict__ vh, float* __restrict__ out, int Nq, int Nk) {
  const int idx = blockIdx.x;
  const int qi = idx % Nq;
  const int tmp = idx / Nq;
  const int h = tmp % NHEAD;
  const int b = tmp / NHEAD;
  __shared__ float sc[960];
  __shared__ float qv[32];
  __shared__ float red[128];
  const int t = threadIdx.x;
  if (t < 32) qv[t] = qh[((size_t)(b * Nq + qi)) * D_MODEL + h * DHEAD + t];
  __syncthreads();
  const float scale = rsqrtf((float)DHEAD);
  for (int k = t; k < Nk; k += 128) {
    const float* kp = &kh[((size_t)(b * Nk + k)) * D_MODEL + h * DHEAD];
    float s = 0.f;
#pragma unroll
    for (int d = 0; d < DHEAD; ++d) s += qv[d] * kp[d];
    sc[k] = s * scale;
  }
  __syncthreads();
  float m = -3.4e38f;
  for (int k = t; k < Nk; k += 128) m = fmaxf(m, sc[k]);
  red[t] = m;
  __syncthreads();
  for (int s = 64; s > 0; s >>= 1) {
    if (t < s) red[t] = fmaxf(red[t], red[t + s]);
    __syncthreads();
  }
  const float mx = red[0];
  __syncthreads();
  float sum = 0.f;
  for (int k = t; k < Nk; k += 128) {
    const float e = expf(sc[k] - mx);
    sc[k] = e;
    sum += e;
  }
  red[t] = sum;
  __syncthreads();
  for (int s = 64; s > 0; s >>= 1) {
    if (t < s) red[t] += red[t + s];
    __syncthreads();
  }
  const float Z = red[0];
  __syncthreads();
  const int grp = t >> 5;
  const int d = t & 31;
  float acc = 0.f;
  for (int k = grp; k < Nk; k += 4)
    acc += sc[k] * vh[((size_t)(b * Nk + k)) * D_MODEL + h * DHEAD + d];
  red[t] = acc;
  __syncthreads();
  if (t < 32) {
    const float r = red[t] + red[t + 32] + red[t + 64] + red[t + 96];
    out[((size_t)(b * Nq + qi)) * D_MODEL + h * DHEAD + t] = r / Z;
  }
}

// ---------------------------------------------------------------------------
// Host side
// ---------------------------------------------------------------------------
struct MsdaP {
  const float *woff, *boff, *waw, *baw, *wv, *bv, *wo, *bo;
};
struct SaP {
  const float *wq, *bq, *wk, *bk, *wv, *bv, *wo, *bo;
};
struct EncP {
  const float *b1, *b2, *ln1g, *ln1b, *ln2g, *ln2b;
  MsdaP m;
  const float *w1, *w2;
};
struct DecP {
  const float *b1, *b2, *ln1g, *ln1b, *ln2g, *ln2b, *ln3g, *ln3b;
  MsdaP m;
  SaP sa;
  const float *w1, *w2;
};

extern "C" void kernel_launch(void* const* d_in, const int* in_sizes, int n_in,
                              void* d_out, int out_size, void* d_ws,
                              size_t ws_size, hipStream_t stream) {
  const float* src = (const float*)d_in[0];
  const float* pos = (const float*)d_in[1];
  const float* refu = (const float*)d_in[2];        // (BS, NQ, 4)
  const float* tgt_embed = (const float*)d_in[3];   // (NQ, D)
  const float* vr = (const float*)d_in[4];          // (BS, NLVL, 2)

  // JAX pytree flatten order: dict keys sorted; tuples/lists in order.
  // params = { dec:[6], dec_norm:(g,b), enc:[6], ref_head:{b1,b2,w1,w2} }
  int ix = 5;
  auto nxt = [&]() { return (const float*)d_in[ix++]; };
  DecP dec[6];
  for (int l = 0; l < 6; ++l) {
    DecP& p = dec[l];
    p.b1 = nxt(); p.b2 = nxt();
    p.ln1g = nxt(); p.ln1b = nxt();
    p.ln2g = nxt(); p.ln2b = nxt();
    p.ln3g = nxt(); p.ln3b = nxt();
    p.m.baw = nxt(); p.m.bo = nxt(); p.m.boff = nxt(); p.m.bv = nxt();
    p.m.waw = nxt(); p.m.wo = nxt(); p.m.woff = nxt(); p.m.wv = nxt();
    p.sa.bk = nxt(); p.sa.bo = nxt(); p.sa.bq = nxt(); p.sa.bv = nxt();
    p.sa.wk = nxt(); p.sa.wo = nxt(); p.sa.wq = nxt(); p.sa.wv = nxt();
    p.w1 = nxt(); p.w2 = nxt();
  }
  const float* dnorm_g = nxt();
  const float* dnorm_b = nxt();
  EncP enc[6];
  for (int l = 0; l < 6; ++l) {
    EncP& p = enc[l];
    p.b1 = nxt(); p.b2 = nxt();
    p.ln1g = nxt(); p.ln1b = nxt();
    p.ln2g = nxt(); p.ln2b = nxt();
    p.m.baw = nxt(); p.m.bo = nxt(); p.m.boff = nxt(); p.m.bv = nxt();
    p.m.waw = nxt(); p.m.wo = nxt(); p.m.woff = nxt(); p.m.wv = nxt();
    p.w1 = nxt(); p.w2 = nxt();
  }
  const float* rb1 = nxt();
  const float* rb2 = nxt();
  const float* rw1 = nxt();
  const float* rw2 = nxt();

  // ---- workspace carve-out (floats) ----
  char* wsp = (char*)d_ws;
  auto alloc = [&](size_t nfl) {
    float* p = (float*)wsp;
    wsp += nfl * sizeof(float);
    return p;
  };
  float* fq    = alloc((size_t)ENC_TOK * D_MODEL);  // q = x+pos; reused as samp
  float* fval  = alloc((size_t)ENC_TOK * D_MODEL);
  float* foff  = alloc((size_t)ENC_TOK * D_MODEL);
  float* faw   = alloc((size_t)ENC_TOK * 128);
  float* ft2   = alloc((size_t)ENC_TOK * D_MODEL);
  float* fmem  = alloc((size_t)ENC_TOK * D_MODEL);
  float* fhid  = alloc((size_t)FFN_CHUNK * DFF_);
  float* frp   = alloc((size_t)BS_ * SUMHW * 2);
  float* frefp = alloc((size_t)DEC_TOK * 4);
  float* fqse  = alloc((size_t)DEC_TOK * 512);
  float* fqpos = alloc((size_t)DEC_TOK * D_MODEL);
  float* ftgt  = alloc((size_t)DEC_TOK * D_MODEL);
  float* fdq   = alloc((size_t)DEC_TOK * D_MODEL);
  float* fqh   = alloc((size_t)DEC_TOK * D_MODEL);
  float* fkh   = alloc((size_t)DEC_TOK * D_MODEL);
  float* fvh   = alloc((size_t)DEC_TOK * D_MODEL);
  float* fattn = alloc((size_t)DEC_TOK * D_MODEL);
  float* fdt2  = alloc((size_t)DEC_TOK * D_MODEL);
  float* fsamp = fq;  // sampling never reads q -> safe alias
  (void)ws_size; (void)n_in; (void)in_sizes; (void)out_size;

  auto gemm = [&](const float* A, const float* B, const float* bias, float* C,
                  int M, int N, int K, int relu) {
    dim3 grid((M + MT - 1) / MT, N / NT);
    if (relu)
      wmma_gemm_bias<1><<<grid, 256, 0, stream>>>(A, B, bias, C, M, N, K);
    else
      wmma_gemm_bias<0><<<grid, 256, 0, stream>>>(A, B, bias, C, M, N, K);
  };
  auto ln = [&](const float* x, const float* res, const float* g,
                const float* b, float* y, int rows) {
    ln_res_kernel<<<rows, 256, 0, stream>>>(x, res, g, b, y);
  };
  auto vadd = [&](const float* a, const float* b, float* c, size_t n) {
    add_kernel<<<(unsigned)((n + 255) / 256), 256, 0, stream>>>(a, b, c, n);
  };
  auto ffn = [&](const float* x, const float* w1, const float* b1,
                 const float* w2, const float* b2, float* outb, int rows) {
    for (int r0 = 0; r0 < rows; r0 += FFN_CHUNK) {
      int mr = rows - r0;
      if (mr > FFN_CHUNK) mr = FFN_CHUNK;
      gemm(x + (size_t)r0 * D_MODEL, w1, b1, fhid, mr, DFF_, D_MODEL, 1);
      gemm(fhid, w2, b2, outb + (size_t)r0 * D_MODEL, mr, D_MODEL, DFF_, 0);
    }
  };

  const size_t enc_elems = (size_t)ENC_TOK * D_MODEL;
  const size_t dec_elems = (size_t)DEC_TOK * D_MODEL;

  // -------------------- Encoder --------------------
  enc_ref_kernel<<<(BS_ * SUMHW + 255) / 256, 256, 0, stream>>>(vr, frp);
  const float* cur = src;
  for (int l = 0; l < 6; ++l) {
    const EncP& p = enc[l];
    vadd(cur, pos, fq, enc_elems);                                   // q = x+pos
    gemm(cur, p.m.wv, p.m.bv, fval, ENC_TOK, D_MODEL, D_MODEL, 0);   // value
    gemm(fq, p.m.woff, p.m.boff, foff, ENC_TOK, D_MODEL, D_MODEL, 0);
    gemm(fq, p.m.waw, p.m.baw, faw, ENC_TOK, 128, D_MODEL, 0);
    softmax32_kernel<<<(ENC_TOK * NHEAD + 255) / 256, 256, 0, stream>>>(
        faw, ENC_TOK * NHEAD);
    msda_sample_kernel<<<ENC_TOK, 256, 0, stream>>>(fval, foff, faw, frp, vr,
                                                    fsamp, SUMHW, 1);
    gemm(fsamp, p.m.wo, p.m.bo, ft2, ENC_TOK, D_MODEL, D_MODEL, 0);
    ln(cur, ft2, p.ln1g, p.ln1b, fmem, ENC_TOK);
    ffn(fmem, p.w1, p.b1, p.w2, p.b2, ft2, ENC_TOK);
    ln(fmem, ft2, p.ln2g, p.ln2b, fmem, ENC_TOK);
    cur = fmem;
  }

  // -------------------- Decoder prep --------------------
  bcast_tgt_kernel<<<DEC_TOK, 256, 0, stream>>>(tgt_embed, ftgt);
  sigmoid_kernel<<<(DEC_TOK * 4 + 255) / 256, 256, 0, stream>>>(refu, frefp,
                                                                DEC_TOK * 4);
  // qpos is layer-invariant (refpoints never updated in this reference)
  sine_embed_kernel<<<DEC_TOK, 512, 0, stream>>>(frefp, vr, fqse);
  gemm(fqse, rw1, rb1, fqh, DEC_TOK, D_MODEL, 512, 1);
  gemm(fqh, rw2, rb2, fqpos, DEC_TOK, D_MODEL, D_MODEL, 0);

  // -------------------- Decoder layers --------------------
  for (int l = 0; l < 6; ++l) {
    const DecP& p = dec[l];
    // self attention
    vadd(ftgt, fqpos, fdq, dec_elems);
    gemm(fdq, p.sa.wq, p.sa.bq, fqh, DEC_TOK, D_MODEL, D_MODEL, 0);
    gemm(fdq, p.sa.wk, p.sa.bk, fkh, DEC_TOK, D_MODEL, D_MODEL, 0);
    gemm(ftgt, p.sa.wv, p.sa.bv, fvh, DEC_TOK, D_MODEL, D_MODEL, 0);
    mha_attn_kernel<<<BS_ * NHEAD * NQ_, 128, 0, stream>>>(fqh, fkh, fvh,
                                                           fattn, NQ_, NQ_);
    gemm(fattn, p.sa.wo, p.sa.bo, fdt2, DEC_TOK, D_MODEL, D_MODEL, 0);
    ln(ftgt, fdt2, p.ln2g, p.ln2b, ftgt, DEC_TOK);
    // cross (deformable) attention over encoder memory
    vadd(ftgt, fqpos, fdq, dec_elems);
    gemm(fmem, p.m.wv, p.m.bv, fval, ENC_TOK, D_MODEL, D_MODEL, 0);
    gemm(fdq, p.m.woff, p.m.boff, foff, DEC_TOK, D_MODEL, D_MODEL, 0);
    gemm(fdq, p.m.waw, p.m.baw, faw, DEC_TOK, 128, D_MODEL, 0);
    softmax32_kernel<<<(DEC_TOK * NHEAD + 255) / 256, 256, 0, stream>>>(
        faw, DEC_TOK * NHEAD);
    msda_sample_kernel<<<DEC_TOK, 256, 0, stream>>>(fval, foff, faw, frefp, vr,
                                                    fsamp, NQ_, 0);
    gemm(fsamp, p.m.wo, p.m.bo, fdt2, DEC_TOK, D_MODEL, D_MODEL, 0);
    ln(ftgt, fdt2, p.ln1g, p.ln1b, ftgt, DEC_TOK);
    // FFN
    ffn(ftgt, p.w1, p.b1, p.w2, p.b2, fdt2, DEC_TOK);
    ln(ftgt, fdt2, p.ln3g, p.ln3b, ftgt, DEC_TOK);
  }

  // final norm -> output (BS, NQ, D) fp32
  ln(ftgt, nullptr, dnorm_g, dnorm_b, (float*)d_out, DEC_TOK);
}